// EFAdLIF_86741159510472
// MI455X (gfx1250) — compile-verified
//
#include <hip/hip_runtime.h>

#define T_STEPS 1000
#define B_SZ 32
#define IN_SZ 512
#define N_SZ 512
#define THR_C 1.0f
#define Q_SC 60.0f

typedef __attribute__((ext_vector_type(2))) float v2f;
typedef __attribute__((ext_vector_type(8))) float v8f;

// ---------------------------------------------------------------------------
// Phase 1: S[row, n] = x[row, :] . weight[n, :] + bias[n],  row = t*B + b
// One wave per 16x16 output tile, K=512 via 128x V_WMMA_F32_16X16X4_F32.
// Written directly into d_out (same [T,B,N] shape).
// ---------------------------------------------------------------------------
__global__ __launch_bounds__(256) void input_proj_kernel(
    const float* __restrict__ x, const float* __restrict__ weight,
    const float* __restrict__ bias, float* __restrict__ out) {
  const int lane = threadIdx.x & 31;
  const int l15  = lane & 15;
  const int h    = (lane >> 4) & 1;
  const int wave = threadIdx.x >> 5;
  const int gw   = blockIdx.x * 8 + wave;   // 0..63999
  const int nt   = gw & 31;                 // 32 n-tiles
  const int mt   = gw >> 5;                 // 2000 row-tiles
  const int row0 = mt * 16;
  const int n0   = nt * 16;

  // A: x[row0 + l15][k],  B: weight[n0 + l15][k];  per reg v: K = 2*h + v
  const float* arow = x      + (size_t)(row0 + l15) * IN_SZ + 2 * h;
  const float* brow = weight + (size_t)(n0   + l15) * IN_SZ + 2 * h;

  v8f acc0 = {}; v8f acc1 = {};
#pragma unroll 8
  for (int ks = 0; ks < 128; ks += 2) {
    v2f av0 = *reinterpret_cast<const v2f*>(arow + 4 * ks);
    v2f bv0 = *reinterpret_cast<const v2f*>(brow + 4 * ks);
    acc0 = __builtin_amdgcn_wmma_f32_16x16x4_f32(false, av0, false, bv0,
                                                 (short)0, acc0, false, false);
    v2f av1 = *reinterpret_cast<const v2f*>(arow + 4 * (ks + 1));
    v2f bv1 = *reinterpret_cast<const v2f*>(brow + 4 * (ks + 1));
    acc1 = __builtin_amdgcn_wmma_f32_16x16x4_f32(false, av1, false, bv1,
                                                 (short)0, acc1, false, false);
  }

  const float bn = bias[n0 + l15];
#pragma unroll
  for (int r = 0; r < 8; ++r) {
    const int row = row0 + r + 8 * h;       // D reg r -> M = r + 8*h
    out[(size_t)row * N_SZ + (n0 + l15)] = acc0[r] + acc1[r] + bn;
  }
}

// ---------------------------------------------------------------------------
// Grid-wide sense-reversing barrier (8 workgroups), short s_sleep backoff.
// ---------------------------------------------------------------------------
__device__ __forceinline__ void grid_barrier(unsigned* cnt, unsigned* gen,
                                             unsigned nwg) {
  __threadfence();            // publish this WG's z stores device-wide
  __syncthreads();
  if (threadIdx.x == 0) {
    unsigned old = __hip_atomic_load(gen, __ATOMIC_RELAXED,
                                     __HIP_MEMORY_SCOPE_AGENT);
    unsigned arrived = __hip_atomic_fetch_add(cnt, 1u, __ATOMIC_ACQ_REL,
                                              __HIP_MEMORY_SCOPE_AGENT);
    if (arrived == nwg - 1u) {
      __hip_atomic_store(cnt, 0u, __ATOMIC_RELAXED, __HIP_MEMORY_SCOPE_AGENT);
      __hip_atomic_store(gen, old + 1u, __ATOMIC_RELEASE,
                         __HIP_MEMORY_SCOPE_AGENT);
    } else {
      while (__hip_atomic_load(gen, __ATOMIC_ACQUIRE,
                               __HIP_MEMORY_SCOPE_AGENT) == old) {
        __builtin_amdgcn_s_sleep(1);   // ~64 clk backoff: fast wake, 8 spinners
      }
    }
  }
  __syncthreads();
  __threadfence();            // acquire: invalidate stale lines before reads
}

__global__ void init_barrier_kernel(unsigned* bar) { bar[0] = 0u; bar[1] = 0u; }

// ---------------------------------------------------------------------------
// Phase 2: persistent sequential scan. 8 WGs x 256 thr = 64 waves; each wave
// owns one 16x16 (b,n) tile for all T steps; u/z/w1/w2 state in registers.
// Per step: rec = z_{t-1} @ R^T via 128 f32 WMMAs; z_{t-1} read from
// d_out[t-1]; S[t] read from d_out[t] (loads hoisted above the K-loop,
// S[t+1] prefetched) and overwritten in place with z_t.
// ---------------------------------------------------------------------------
__global__ __launch_bounds__(256) void recurrent_scan_kernel(
    const float* __restrict__ z0,  const float* __restrict__ u0,
    const float* __restrict__ w10, const float* __restrict__ w20,
    const float* __restrict__ recur,
    const float* __restrict__ decay_u, const float* __restrict__ decay_w1,
    const float* __restrict__ decay_w2,
    const float* __restrict__ a1p, const float* __restrict__ a2p,
    const float* __restrict__ b1p, const float* __restrict__ b2p,
    float* __restrict__ out, unsigned* __restrict__ bar) {
  const int lane = threadIdx.x & 31;
  const int l15  = lane & 15;
  const int h    = (lane >> 4) & 1;
  const int wave = threadIdx.x >> 5;
  const int gw   = blockIdx.x * 8 + wave;   // 0..63
  const int mt   = gw & 1;                  // 2 batch tiles
  const int nt   = gw >> 1;                 // 32 neuron tiles
  const int b0   = mt * 16;
  const int n0   = nt * 16;
  const int n    = n0 + l15;

  // per-neuron parameters (depend on n only -> one scalar per lane)
  const float du  = decay_u[n];
  const float dw1 = decay_w1[n];
  const float dw2 = decay_w2[n];
  const float A1  = a1p[n], A2 = a2p[n];
  const float B1  = b1p[n], B2 = b2p[n];

  // step-invariant epilogue indices: D reg r -> (b = b0 + r + 8h, n)
  int eidx[8];
#pragma unroll
  for (int r = 0; r < 8; ++r)
    eidx[r] = (b0 + r + 8 * h) * N_SZ + n;

  // per-(b,n) state, register resident for all 1000 steps
  float u[8], z[8], w1[8], w2[8];
#pragma unroll
  for (int r = 0; r < 8; ++r) {
    u[r] = u0[eidx[r]]; z[r] = z0[eidx[r]];
    w1[r] = w10[eidx[r]]; w2[r] = w20[eidx[r]];
  }

  // B fragments: R[n][k] (L2-resident, step-invariant)
  const float* brow = recur + (size_t)n * N_SZ + 2 * h;

  for (int t = 0; t < T_STEPS; ++t) {
    const float* zsrc = (t == 0) ? z0 : (out + (size_t)(t - 1) * B_SZ * N_SZ);
    const float* arow = zsrc + (size_t)(b0 + l15) * N_SZ + 2 * h;
    float* orow = out + (size_t)t * B_SZ * N_SZ;

    // Hoist S[t] loads: overlap their latency with the 128-WMMA K-loop.
    float s[8];
#pragma unroll
    for (int r = 0; r < 8; ++r) s[r] = orow[eidx[r]];

    // Prefetch next step's S tile into cache (global_prefetch_b8),
    // overlapped with the GEMM and the barrier wait.
    if (t + 1 < T_STEPS) {
      const float* nrow = orow + (size_t)B_SZ * N_SZ;
#pragma unroll
      for (int r = 0; r < 8; ++r) __builtin_prefetch(nrow + eidx[r], 0, 1);
    }

    v8f acc0 = {}; v8f acc1 = {};
#pragma unroll 8
    for (int ks = 0; ks < 128; ks += 2) {
      v2f av0 = *reinterpret_cast<const v2f*>(arow + 4 * ks);
      v2f bv0 = *reinterpret_cast<const v2f*>(brow + 4 * ks);
      acc0 = __builtin_amdgcn_wmma_f32_16x16x4_f32(false, av0, false, bv0,
                                                   (short)0, acc0, false, false);
      v2f av1 = *reinterpret_cast<const v2f*>(arow + 4 * (ks + 1));
      v2f bv1 = *reinterpret_cast<const v2f*>(brow + 4 * (ks + 1));
      acc1 = __builtin_amdgcn_wmma_f32_16x16x4_f32(false, av1, false, bv1,
                                                   (short)0, acc1, false, false);
    }

#pragma unroll
    for (int r = 0; r < 8; ++r) {
      const float soma  = s[r] + acc0[r] + acc1[r];      // S[t] + z@R^T
      const float u_new = du * u[r] + (1.0f - du) * (soma - w1[r] - w2[r]);
      const float spike = (u_new - THR_C > 0.0f) ? 1.0f : 0.0f;
      // NOTE: w2 update intentionally uses w1 (faithful to reference),
      // and both use pre-update u, z.
      const float w1n = dw1 * w1[r] + (1.0f - dw1) * (A1 * u[r] + B1 * z[r]) * Q_SC;
      const float w2n = dw2 * w1[r] + (1.0f - dw2) * (A2 * u[r] + B2 * z[r]) * Q_SC;
      u[r]  = u_new * (1.0f - spike);
      z[r]  = spike;
      w1[r] = w1n;
      w2[r] = w2n;
      orow[eidx[r]] = spike;                             // overwrite S with z_t
    }

    grid_barrier(bar, bar + 1, gridDim.x);
  }
}

// ---------------------------------------------------------------------------
extern "C" void kernel_launch(void* const* d_in, const int* in_sizes, int n_in,
                              void* d_out, int out_size, void* d_ws,
                              size_t ws_size, hipStream_t stream) {
  const float* x      = (const float*)d_in[0];
  const float* u0     = (const float*)d_in[1];
  const float* z0     = (const float*)d_in[2];
  const float* w10    = (const float*)d_in[3];
  const float* w20    = (const float*)d_in[4];
  const float* weight = (const float*)d_in[5];
  const float* bias   = (const float*)d_in[6];
  const float* recur  = (const float*)d_in[7];
  const float* du     = (const float*)d_in[8];
  const float* dw1    = (const float*)d_in[9];
  const float* dw2    = (const float*)d_in[10];
  const float* a1     = (const float*)d_in[11];
  const float* a2     = (const float*)d_in[12];
  const float* b1     = (const float*)d_in[13];
  const float* b2     = (const float*)d_in[14];
  float* out    = (float*)d_out;
  unsigned* bar = (unsigned*)d_ws;

  init_barrier_kernel<<<1, 1, 0, stream>>>(bar);
  // 2000 row-tiles * 32 n-tiles = 64000 tiles, 8 waves (tiles) per block
  input_proj_kernel<<<8000, 256, 0, stream>>>(x, weight, bias, out);
  recurrent_scan_kernel<<<8, 256, 0, stream>>>(z0, u0, w10, w20, recur,
                                               du, dw1, dw2, a1, a2, b1, b2,
                                               out, bar);
}